// GCN_49237505081890
// MI455X (gfx1250) — compile-verified
//
#include <hip/hip_runtime.h>
#include <math.h>

#define NN  16000
#define KK  16
#define FIN 32
#define HH  64

typedef float v2f __attribute__((ext_vector_type(2)));
typedef float v8f __attribute__((ext_vector_type(8)));

// ---------------------------------------------------------------------------
// 1) Brute-force KNN: one thread per center, coordinates streamed via LDS in
//    16KB chunks. Top-16 kept as a fully unrolled register insertion list.
// ---------------------------------------------------------------------------
__global__ __launch_bounds__(256) void knn_kernel(const float* __restrict__ c,
                                                  int* __restrict__ nidx,
                                                  float* __restrict__ ndist) {
  __shared__ float2 sh[2048];
  const int i = blockIdx.x * 256 + threadIdx.x;
  const bool active = (i < NN);
  float xi = 0.0f, yi = 0.0f;
  if (active) { xi = c[2 * i]; yi = c[2 * i + 1]; }

  float bd[KK];
  int   bi[KK];
#pragma unroll
  for (int t = 0; t < KK; ++t) { bd[t] = 3.0e38f; bi[t] = 0; }

  for (int base = 0; base < NN; base += 2048) {
    const int cnt = min(2048, NN - base);
    __syncthreads();
    for (int t = threadIdx.x; t < cnt; t += 256)
      sh[t] = ((const float2*)c)[base + t];
    __syncthreads();
    if (active) {
      for (int p = 0; p < cnt; ++p) {
        const float dx = xi - sh[p].x;
        const float dy = yi - sh[p].y;
        const float d2 = dx * dx + dy * dy;
        const int   j  = base + p;
        if (d2 < bd[KK - 1] && j != i) {   // rare branch: insertion
          float d = d2; int id = j;
#pragma unroll
          for (int t = 0; t < KK; ++t) {
            if (d < bd[t]) {
              float td = bd[t]; int ti = bi[t];
              bd[t] = d; bi[t] = id;
              d = td; id = ti;
            }
          }
        }
      }
    }
  }
  if (active) {
#pragma unroll
    for (int t = 0; t < KK; ++t) {
      nidx[i * KK + t]  = bi[t];
      ndist[i * KK + t] = sqrtf(bd[t]);
    }
  }
}

// ---------------------------------------------------------------------------
// 2) Global min/max of all N*K distances: single 1024-thread block reduction.
// ---------------------------------------------------------------------------
__global__ __launch_bounds__(1024) void minmax_kernel(const float* __restrict__ d,
                                                      float* __restrict__ mm) {
  __shared__ float smn[1024];
  __shared__ float smx[1024];
  float mn = 3.0e38f, mx = -3.0e38f;
  for (int t = threadIdx.x; t < NN * KK; t += 1024) {
    const float v = d[t];
    mn = fminf(mn, v);
    mx = fmaxf(mx, v);
  }
  smn[threadIdx.x] = mn;
  smx[threadIdx.x] = mx;
  __syncthreads();
  for (int s = 512; s > 0; s >>= 1) {
    if (threadIdx.x < s) {
      smn[threadIdx.x] = fminf(smn[threadIdx.x], smn[threadIdx.x + s]);
      smx[threadIdx.x] = fmaxf(smx[threadIdx.x], smx[threadIdx.x + s]);
    }
    __syncthreads();
  }
  if (threadIdx.x == 0) { mm[0] = smn[0]; mm[1] = smx[0]; }
}

// ---------------------------------------------------------------------------
// 3) Edge weights + symmetric-norm factors. Edges of node i are contiguous
//    [i*K, i*K+K) with dst==i, so degree needs no atomics.
// ---------------------------------------------------------------------------
__global__ void ewdeg_kernel(const float* __restrict__ ndist,
                             const float* __restrict__ mm,
                             float* __restrict__ ew,
                             float* __restrict__ dis) {
  const int i = blockIdx.x * blockDim.x + threadIdx.x;
  if (i >= NN) return;
  const float mn  = mm[0];
  const float mx  = mm[1];
  const float inv = 1.0f / (mx - mn);
  float deg = 1.0f;                       // self loop weight 1
#pragma unroll
  for (int t = 0; t < KK; ++t) {
    const float w = (mx - ndist[i * KK + t]) * inv;
    ew[i * KK + t] = w;
    deg += w;
  }
  dis[i] = rsqrtf(deg);                   // deg >= 1 always
}

// ---------------------------------------------------------------------------
// 4) Dense GEMM Y[N,64] = X[N,KDIM] @ W[KDIM,64] via V_WMMA_F32_16X16X4_F32.
//    One wave per 16x16 output tile; 1000 row tiles x 4 col tiles, no tails,
//    EXEC all-1s per wave. A: lane=M, (reg, lane-half)=K; B mirrored lane=N;
//    C/D: reg r holds M=r (lanes 0-15) and M=r+8 (lanes 16-31), N=lane&15.
// ---------------------------------------------------------------------------
template <int KDIM>
__global__ __launch_bounds__(256) void gemm_wmma(const float* __restrict__ X,
                                                 const float* __restrict__ W,
                                                 float* __restrict__ Y) {
  const int wave = blockIdx.x * 8 + (threadIdx.x >> 5);
  const int tn = wave & 3;        // 64/16 = 4 column tiles
  const int tm = wave >> 2;       // 16000/16 = 1000 row tiles
  if (tm >= NN / 16) return;      // uniform per wave (never taken: exact grid)

  const int lane = threadIdx.x & 31;
  const int half = lane >> 4;     // K sub-pair select
  const int lm   = lane & 15;

  v8f acc = {};
  const float* xrow = X + (size_t)(tm * 16 + lm) * KDIM;
#pragma unroll
  for (int k0 = 0; k0 < KDIM; k0 += 4) {
    const int ka = k0 + 2 * half;
    v2f a, b;
    a.x = xrow[ka];
    a.y = xrow[ka + 1];
    b.x = W[(ka + 0) * HH + tn * 16 + lm];
    b.y = W[(ka + 1) * HH + tn * 16 + lm];
    acc = __builtin_amdgcn_wmma_f32_16x16x4_f32(false, a, false, b,
                                                (short)0, acc, false, false);
  }
#pragma unroll
  for (int r = 0; r < 8; ++r) {
    const int m = r + 8 * half;
    Y[(size_t)(tm * 16 + m) * HH + tn * 16 + lm] = acc[r];
  }
}

// ---------------------------------------------------------------------------
// 5) GCN aggregation + bias + ReLU.
//    out_i = dis_i*( sum_k dis_j*ew_k*xw_j + dis_i*xw_i ) + b, then ReLU.
//    One thread per (node, feature).
// ---------------------------------------------------------------------------
__global__ void agg_kernel(const float* __restrict__ xw,
                           const int* __restrict__ nidx,
                           const float* __restrict__ ew,
                           const float* __restrict__ dis,
                           const float* __restrict__ bias,
                           float* __restrict__ h) {
  const int gid = blockIdx.x * blockDim.x + threadIdx.x;
  const int i = gid >> 6;
  const int f = gid & 63;
  if (i >= NN) return;
  const float di = dis[i];
  float acc = di * xw[(size_t)i * HH + f];          // self-loop term
#pragma unroll
  for (int t = 0; t < KK; ++t) {
    const int   j = nidx[i * KK + t];
    const float w = ew[i * KK + t] * dis[j];
    acc = fmaf(w, xw[(size_t)j * HH + f], acc);
  }
  const float o = fmaf(di, acc, bias[f]);
  h[(size_t)i * HH + f] = fmaxf(o, 0.0f);
}

// ---------------------------------------------------------------------------
// 6) Final fc: out[i] = h2[i,:] . Wfc + bfc  (F_OUT = 1)
// ---------------------------------------------------------------------------
__global__ void fc_kernel(const float* __restrict__ h,
                          const float* __restrict__ Wfc,
                          const float* __restrict__ bfc,
                          float* __restrict__ out) {
  const int i = blockIdx.x * blockDim.x + threadIdx.x;
  if (i >= NN) return;
  float acc = bfc[0];
#pragma unroll
  for (int f = 0; f < HH; ++f)
    acc = fmaf(h[(size_t)i * HH + f], Wfc[f], acc);
  out[i] = acc;
}

// ---------------------------------------------------------------------------
extern "C" void kernel_launch(void* const* d_in, const int* in_sizes, int n_in,
                              void* d_out, int out_size, void* d_ws, size_t ws_size,
                              hipStream_t stream) {
  (void)in_sizes; (void)n_in; (void)out_size; (void)ws_size;
  const float* x   = (const float*)d_in[0];
  const float* c   = (const float*)d_in[1];
  const float* W1  = (const float*)d_in[2];
  const float* b1  = (const float*)d_in[3];
  const float* W2  = (const float*)d_in[4];
  const float* b2  = (const float*)d_in[5];
  const float* Wfc = (const float*)d_in[6];
  const float* bfc = (const float*)d_in[7];
  float* out = (float*)d_out;

  char* ws = (char*)d_ws;
  size_t off = 0;
  int*   nidx  = (int*)(ws + off);   off += (size_t)NN * KK * sizeof(int);
  float* ndist = (float*)(ws + off); off += (size_t)NN * KK * sizeof(float);
  float* ew    = (float*)(ws + off); off += (size_t)NN * KK * sizeof(float);
  float* dis   = (float*)(ws + off); off += (size_t)NN * sizeof(float);
  float* mm    = (float*)(ws + off); off += 16;
  float* bufA  = (float*)(ws + off); off += (size_t)NN * HH * sizeof(float);
  float* bufB  = (float*)(ws + off); off += (size_t)NN * HH * sizeof(float);

  knn_kernel<<<(NN + 255) / 256, 256, 0, stream>>>(c, nidx, ndist);
  minmax_kernel<<<1, 1024, 0, stream>>>(ndist, mm);
  ewdeg_kernel<<<(NN + 255) / 256, 256, 0, stream>>>(ndist, mm, ew, dis);

  // layer 1: xw1 = x @ W1  (K=32) -> bufA; h1 = relu(agg(xw1)+b1) -> bufB
  gemm_wmma<FIN><<<(NN / 16) * 4 / 8, 256, 0, stream>>>(x, W1, bufA);
  agg_kernel<<<(NN * HH + 255) / 256, 256, 0, stream>>>(bufA, nidx, ew, dis, b1, bufB);

  // layer 2: xw2 = h1 @ W2 (K=64) -> bufA; h2 = relu(agg(xw2)+b2) -> bufB
  gemm_wmma<HH><<<(NN / 16) * 4 / 8, 256, 0, stream>>>(bufB, W2, bufA);
  agg_kernel<<<(NN * HH + 255) / 256, 256, 0, stream>>>(bufA, nidx, ew, dis, b2, bufB);

  // fc head
  fc_kernel<<<(NN + 255) / 256, 256, 0, stream>>>(bufB, Wfc, bfc, out);
}